// Classify_graph_gc_128849019551
// MI455X (gfx1250) — compile-verified
//
#include <hip/hip_runtime.h>
#include <math.h>

#define NN 100000
#define EE 1600000
#define HH 128
#define GG 1000
#define CC 10

typedef __attribute__((ext_vector_type(2))) float v2f;
typedef __attribute__((ext_vector_type(8))) float v8f;

// ---------- degree / normalization ----------
__global__ void k_init_deg(float* deg) {
    int i = blockIdx.x * blockDim.x + threadIdx.x;
    if (i < NN) deg[i] = 1.0f;  // self-loop contribution
}

__global__ void k_count_deg(const int* __restrict__ ei, float* deg) {
    int e = blockIdx.x * blockDim.x + threadIdx.x;
    if (e < EE) {
        int d = ei[EE + e];  // dst row of edge_index
        unsafeAtomicAdd(&deg[d], 1.0f);
    }
}

__global__ void k_dinv(float* deg) {
    int i = blockIdx.x * blockDim.x + threadIdx.x;
    if (i < NN) deg[i] = rsqrtf(deg[i]);  // deg >= 1 always
}

// ---------- GEMM: H = X(N x 128) * W(128 x 128), fp32 WMMA ----------
// block = 256 threads (8 waves). Each block computes 16 rows x 128 cols.
// Wave w handles column tile w*16. K-loop: 32 chained V_WMMA_F32_16X16X4_F32.
__global__ void __launch_bounds__(256) k_gemm128(const float* __restrict__ X,
                                                 const float* __restrict__ W,
                                                 float* __restrict__ Hout) {
    int lane = threadIdx.x & 31;
    int wv   = threadIdx.x >> 5;   // 0..7 -> column tile
    int half = lane >> 4;          // 0/1
    int l    = lane & 15;
    int row0 = blockIdx.x << 4;    // 16 rows per block (N divisible by 16)
    int col0 = wv << 4;

    const float* xrow = X + (size_t)(row0 + l) * HH;
    v8f c = {};
    for (int kb = 0; kb < HH; kb += 4) {
        int ka = kb + 2 * half;                      // K pair this half-wave owns
        v2f a = *(const v2f*)(xrow + ka);            // A[M=l][ka], A[M=l][ka+1]
        v2f b;
        b.x = W[(size_t)ka       * HH + col0 + l];   // B[ka][col0+l]
        b.y = W[(size_t)(ka + 1) * HH + col0 + l];   // B[ka+1][col0+l]
        c = __builtin_amdgcn_wmma_f32_16x16x4_f32(false, a, false, b,
                                                  (short)0, c, false, false);
    }
#pragma unroll
    for (int v = 0; v < 8; ++v) {
        Hout[(size_t)(row0 + v + 8 * half) * HH + col0 + l] = c[v];
    }
}

// ---------- aggregation: acc = dinv^2 * h (self-loop), then edge scatter-add ----------
__global__ void k_agg_init(const float* __restrict__ h, const float* __restrict__ dinv,
                           float* __restrict__ acc) {
    int i = blockIdx.x * blockDim.x + threadIdx.x;
    if (i < NN * HH) {
        int n = i >> 7;
        float w = dinv[n];
        acc[i] = w * w * h[i];
    }
}

// One wave per edge; lane handles 4 features (float4 gather, 4x f32 atomic add).
__global__ void k_agg_edges(const int* __restrict__ ei, const float* __restrict__ dinv,
                            const float* __restrict__ h, float* __restrict__ acc) {
    long long tid = (long long)blockIdx.x * blockDim.x + threadIdx.x;
    int e    = (int)(tid >> 5);
    int lane = (int)(tid & 31);
    if (e < EE) {
        int s = ei[e];
        int d = ei[EE + e];
        float w = dinv[s] * dinv[d];
        float4 v = *((const float4*)(h + (size_t)s * HH) + lane);
        float* ap = acc + (size_t)d * HH + lane * 4;
        unsafeAtomicAdd(ap + 0, w * v.x);
        unsafeAtomicAdd(ap + 1, w * v.y);
        unsafeAtomicAdd(ap + 2, w * v.z);
        unsafeAtomicAdd(ap + 3, w * v.w);
    }
}

__global__ void k_bias_elu(float* __restrict__ x, const float* __restrict__ b) {
    int i = blockIdx.x * blockDim.x + threadIdx.x;
    if (i < NN * HH) {
        float v = x[i] + b[i & (HH - 1)];
        x[i] = v > 0.0f ? v : (__expf(v) - 1.0f);
    }
}

// ---------- segment-max pooling via monotone-uint atomicMax ----------
__device__ __forceinline__ unsigned fenc(float v) {
    unsigned u = __float_as_uint(v);
    return (u & 0x80000000u) ? ~u : (u | 0x80000000u);
}
__device__ __forceinline__ float fdec(unsigned k) {
    unsigned u = (k & 0x80000000u) ? (k ^ 0x80000000u) : ~k;
    return __uint_as_float(u);
}

__global__ void k_pool_init(unsigned* pk) {
    int i = blockIdx.x * blockDim.x + threadIdx.x;
    if (i < GG * HH) pk[i] = fenc(-INFINITY);
}

__global__ void k_pool_max(const float* __restrict__ x, const int* __restrict__ batch,
                           unsigned* pk) {
    int i = blockIdx.x * blockDim.x + threadIdx.x;
    if (i < NN * HH) {
        int n = i >> 7, f = i & (HH - 1);
        int g = batch[n];
        atomicMax(&pk[g * HH + f], fenc(x[i]));
    }
}

// ---------- head: logits = pooled @ Wl + bl, softmax ----------
__global__ void __launch_bounds__(128) k_head(const unsigned* __restrict__ pk,
                                              const float* __restrict__ Wl,
                                              const float* __restrict__ bl,
                                              float* __restrict__ out) {
    __shared__ float p[HH];
    __shared__ float lg[CC];
    int g = blockIdx.x;
    int t = threadIdx.x;
    float v = fdec(pk[g * HH + t]);
    if (!isfinite(v)) v = 0.0f;  // empty-segment guard
    p[t] = v;
    __syncthreads();
    if (t < CC) {
        float s = bl[t];
        for (int f = 0; f < HH; ++f) s += p[f] * Wl[f * CC + t];
        lg[t] = s;
    }
    __syncthreads();
    if (t == 0) {
        float m = lg[0];
        for (int c2 = 1; c2 < CC; ++c2) m = fmaxf(m, lg[c2]);
        float e[CC];
        float sum = 0.0f;
        for (int c2 = 0; c2 < CC; ++c2) { e[c2] = __expf(lg[c2] - m); sum += e[c2]; }
        float inv = 1.0f / sum;
        for (int c2 = 0; c2 < CC; ++c2) out[g * CC + c2] = e[c2] * inv;
    }
}

extern "C" void kernel_launch(void* const* d_in, const int* in_sizes, int n_in,
                              void* d_out, int out_size, void* d_ws, size_t ws_size,
                              hipStream_t stream) {
    const float* x      = (const float*)d_in[0];
    const int*   ei     = (const int*)d_in[1];      // [2, E] row-major
    const int*   batch  = (const int*)d_in[2];
    const float* W[3]   = {(const float*)d_in[3], (const float*)d_in[5], (const float*)d_in[7]};
    const float* B[3]   = {(const float*)d_in[4], (const float*)d_in[6], (const float*)d_in[8]};
    const float* Wl     = (const float*)d_in[9];
    const float* bl     = (const float*)d_in[10];
    float* out = (float*)d_out;

    // workspace layout (floats): dinv[NN] | poolkeys[GG*HH] | bufA[NN*HH] | bufB[NN*HH]
    float*    ws   = (float*)d_ws;
    float*    dinv = ws;
    unsigned* pk   = (unsigned*)(ws + NN);
    float*    bufA = ws + NN + GG * HH;          // 16B-aligned (228000 floats)
    float*    bufB = bufA + (size_t)NN * HH;

    // normalization coefficients
    k_init_deg <<<(NN + 255) / 256, 256, 0, stream>>>(dinv);
    k_count_deg<<<(EE + 255) / 256, 256, 0, stream>>>(ei, dinv);
    k_dinv     <<<(NN + 255) / 256, 256, 0, stream>>>(dinv);

    // three GCN layers, ping-pong between bufA (h) and bufB (acc / next x)
    const float* xin = x;
    for (int layer = 0; layer < 3; ++layer) {
        k_gemm128  <<<NN / 16, 256, 0, stream>>>(xin, W[layer], bufA);
        k_agg_init <<<(NN * HH + 255) / 256, 256, 0, stream>>>(bufA, dinv, bufB);
        k_agg_edges<<<(int)(((long long)EE * 32 + 255) / 256), 256, 0, stream>>>(ei, dinv, bufA, bufB);
        k_bias_elu <<<(NN * HH + 255) / 256, 256, 0, stream>>>(bufB, B[layer]);
        xin = bufB;
    }

    // pooling + head
    k_pool_init<<<(GG * HH + 255) / 256, 256, 0, stream>>>(pk);
    k_pool_max <<<(NN * HH + 255) / 256, 256, 0, stream>>>(bufB, batch, pk);
    k_head     <<<GG, HH, 0, stream>>>(pk, Wl, bl, out);
}